// GraphConvLayer_77043123356296
// MI455X (gfx1250) — compile-verified
//
#include <hip/hip_runtime.h>
#include <hip/hip_bf16.h>

typedef __attribute__((ext_vector_type(2))) float v2f;
typedef __attribute__((ext_vector_type(8))) float v8f;

#define D 512
#define NEG_SLOPE 0.01f
#define TILE_M 128
#define TILE_N 128
#define TILE_K 16
#define LDA 17    // 16 + 1 pad (kills 16-stride bank conflicts on A-frag reads)
#define LDB 129   // 128 + 1 pad

// ---------------- norm precompute ----------------

__global__ void k_init_deg(float* __restrict__ deg, int n) {
    int i = blockIdx.x * blockDim.x + threadIdx.x;
    if (i < n) deg[i] = 2.0f;   // improved=True self-loop fill
}

__global__ void k_accum_deg(const int* __restrict__ col, const float* __restrict__ w,
                            float* __restrict__ deg, int e) {
    int i = blockIdx.x * blockDim.x + threadIdx.x;
    if (i < e)
        (void)__hip_atomic_fetch_add(&deg[col[i]], w[i],
                                     __ATOMIC_RELAXED, __HIP_MEMORY_SCOPE_AGENT);
}

__global__ void k_rsqrt_deg(float* __restrict__ deg, int n) {
    int i = blockIdx.x * blockDim.x + threadIdx.x;
    if (i < n) {
        float d = deg[i];
        deg[i] = d > 0.0f ? rsqrtf(d) : 0.0f;   // deg -> deg^{-1/2} in place
    }
}

__global__ void k_edge_norm(const int* __restrict__ row, const int* __restrict__ col,
                            const float* __restrict__ w, const float* __restrict__ dis,
                            float* __restrict__ nrm, int e) {
    int i = blockIdx.x * blockDim.x + threadIdx.x;
    if (i < e) nrm[i] = dis[row[i]] * w[i] * dis[col[i]];
}

// ---------------- per-conv aggregation in input space ----------------

// agg[n,:] = (2 * dis[n]^2) * x[n,:]   (self-loop contribution; also zero-inits agg)
__global__ void k_init_agg(const float* __restrict__ x, const float* __restrict__ dis,
                           float* __restrict__ agg, int nvec4) {
    int i = blockIdx.x * blockDim.x + threadIdx.x;   // float4 index
    if (i < nvec4) {
        int node = i >> 7;                            // 128 float4 per row of 512
        float s = dis[node];
        s = 2.0f * s * s;
        const float4 v = ((const float4*)x)[i];
        float4 o;
        o.x = s * v.x; o.y = s * v.y; o.z = s * v.z; o.w = s * v.w;
        ((float4*)agg)[i] = o;
    }
}

// one 128-thread block per edge: agg[col,:] += norm * x[row,:]
__global__ __launch_bounds__(128)
void k_scatter(const int* __restrict__ row, const int* __restrict__ col,
               const float* __restrict__ nrm, const float* __restrict__ x,
               float* __restrict__ agg) {
    int e = blockIdx.x;
    int t = threadIdx.x;                 // 0..127, 4 floats each
    int r = row[e];
    int c = col[e];
    float w = nrm[e];
    const float4 v = *(const float4*)(x + (size_t)r * D + t * 4);
    float* dst = agg + (size_t)c * D + t * 4;
    (void)__hip_atomic_fetch_add(dst + 0, w * v.x, __ATOMIC_RELAXED, __HIP_MEMORY_SCOPE_AGENT);
    (void)__hip_atomic_fetch_add(dst + 1, w * v.y, __ATOMIC_RELAXED, __HIP_MEMORY_SCOPE_AGENT);
    (void)__hip_atomic_fetch_add(dst + 2, w * v.z, __ATOMIC_RELAXED, __HIP_MEMORY_SCOPE_AGENT);
    (void)__hip_atomic_fetch_add(dst + 3, w * v.w, __ATOMIC_RELAXED, __HIP_MEMORY_SCOPE_AGENT);
}

// ---------------- fused WMMA GEMM: out = leakyrelu(agg @ W + x @ Wres) ----------------
// Virtual K = 1024: first 512 uses (agg, W), second 512 uses (x, Wres).
// Block tile 128x128, 8 waves (wave32) arranged 2x4, each wave 64x32 (4x2 wmma tiles).

__global__ __launch_bounds__(256)
void k_gemm_fused(const float* __restrict__ agg, const float* __restrict__ x,
                  const float* __restrict__ W, const float* __restrict__ Wres,
                  float* __restrict__ out, int nrows) {
    __shared__ float As[TILE_M * LDA];
    __shared__ float Bs[TILE_K * LDB];

    const int tid   = threadIdx.x;
    const int lane  = tid & 31;
    const int wid   = tid >> 5;
    const int half  = lane >> 4;      // 0 or 1 (wave32 half)
    const int l     = lane & 15;
    const int gm0   = blockIdx.x * TILE_M;
    const int gn0   = blockIdx.y * TILE_N;
    const int waveM = wid >> 2;       // 0..1 -> 64-row strip
    const int waveN = wid & 3;        // 0..3 -> 32-col strip

    v8f acc[4][2];
#pragma unroll
    for (int mi = 0; mi < 4; ++mi)
#pragma unroll
        for (int ni = 0; ni < 2; ++ni)
            acc[mi][ni] = (v8f){0.f, 0.f, 0.f, 0.f, 0.f, 0.f, 0.f, 0.f};

    for (int ks = 0; ks < 1024 / TILE_K; ++ks) {
        const int kg = ks * TILE_K;
        const float* __restrict__ Ap = (kg < D) ? agg : x;
        const float* __restrict__ Bp = (kg < D) ? W : Wres;
        const int kk = kg & (D - 1);

        // Load A tile: 128 rows x 16 cols = 512 float4 slots, 2 per thread
#pragma unroll
        for (int it = 0; it < 2; ++it) {
            int s  = tid + it * 256;
            int r  = s >> 2;
            int c4 = (s & 3) * 4;
            int gr = gm0 + r;
            float4 v = make_float4(0.f, 0.f, 0.f, 0.f);
            if (gr < nrows) v = *(const float4*)(Ap + (size_t)gr * D + kk + c4);
            float* dst = &As[r * LDA + c4];
            dst[0] = v.x; dst[1] = v.y; dst[2] = v.z; dst[3] = v.w;
        }
        // Load B tile: 16 rows x 128 cols = 512 float4 slots, 2 per thread
#pragma unroll
        for (int it = 0; it < 2; ++it) {
            int s  = tid + it * 256;
            int r  = s >> 5;
            int c4 = (s & 31) * 4;
            float4 v = *(const float4*)(Bp + (size_t)(kk + r) * D + gn0 + c4);
            float* dst = &Bs[r * LDB + c4];
            dst[0] = v.x; dst[1] = v.y; dst[2] = v.z; dst[3] = v.w;
        }
        __syncthreads();

#pragma unroll
        for (int k0 = 0; k0 < TILE_K; k0 += 4) {
            v2f a[4], b[2];
#pragma unroll
            for (int mi = 0; mi < 4; ++mi) {
                // A frag 16x4 f32: lanes 0-15 hold K={k0,k0+1}, lanes 16-31 K={k0+2,k0+3}
                const float* p = &As[(waveM * 64 + mi * 16 + l) * LDA + k0 + 2 * half];
                a[mi].x = p[0];
                a[mi].y = p[1];
            }
#pragma unroll
            for (int ni = 0; ni < 2; ++ni) {
                // B frag 4x16 f32, mirrored K split across wave halves
                const float* p = &Bs[(k0 + 2 * half) * LDB + waveN * 32 + ni * 16 + l];
                b[ni].x = p[0];
                b[ni].y = p[LDB];
            }
#pragma unroll
            for (int mi = 0; mi < 4; ++mi)
#pragma unroll
                for (int ni = 0; ni < 2; ++ni)
                    acc[mi][ni] = __builtin_amdgcn_wmma_f32_16x16x4_f32(
                        false, a[mi], false, b[ni], (short)0, acc[mi][ni], false, false);
        }
        __syncthreads();
    }

    // Epilogue: leaky_relu, store. C layout: VGPR v -> M = v + 8*half, N = l.
#pragma unroll
    for (int mi = 0; mi < 4; ++mi) {
#pragma unroll
        for (int ni = 0; ni < 2; ++ni) {
            int gcol = gn0 + waveN * 32 + ni * 16 + l;
#pragma unroll
            for (int v = 0; v < 8; ++v) {
                int grow = gm0 + waveM * 64 + mi * 16 + v + 8 * half;
                if (grow < nrows) {
                    float val = acc[mi][ni][v];
                    out[(size_t)grow * D + gcol] = val >= 0.f ? val : NEG_SLOPE * val;
                }
            }
        }
    }
}

// ---------------- launcher ----------------

extern "C" void kernel_launch(void* const* d_in, const int* in_sizes, int n_in,
                              void* d_out, int out_size, void* d_ws, size_t ws_size,
                              hipStream_t stream) {
    const float* x0  = (const float*)d_in[0];
    const float* x1  = (const float*)d_in[1];
    const float* x2  = (const float*)d_in[2];
    const int*   adj = (const int*)d_in[3];     // [2, E]: row 0 = src, row 1 = dst
    const float* aw  = (const float*)d_in[4];
    const float* W   = (const float*)d_in[5];
    const float* Wr  = (const float*)d_in[6];
    float*       out = (float*)d_out;

    const int Nn = in_sizes[0] / D;
    const int E  = in_sizes[4];
    const int* rowp = adj;
    const int* colp = adj + E;

    // workspace layout: agg [Nn*D] | deg->dis [Nn] | norm [E]
    float* agg = (float*)d_ws;
    float* deg = agg + (size_t)Nn * D;
    float* nrm = deg + Nn;

    const int T = 256;
    k_init_deg<<<(Nn + T - 1) / T, T, 0, stream>>>(deg, Nn);
    k_accum_deg<<<(E + T - 1) / T, T, 0, stream>>>(colp, aw, deg, E);
    k_rsqrt_deg<<<(Nn + T - 1) / T, T, 0, stream>>>(deg, Nn);
    k_edge_norm<<<(E + T - 1) / T, T, 0, stream>>>(rowp, colp, aw, deg, nrm, E);

    const float* xs[3] = {x0, x1, x2};
    dim3 ggrid((Nn + TILE_M - 1) / TILE_M, D / TILE_N);
    const int nvec4 = Nn * (D / 4);
    for (int i = 0; i < 3; ++i) {
        k_init_agg<<<(nvec4 + T - 1) / T, T, 0, stream>>>(xs[i], deg, agg, nvec4);
        k_scatter<<<E, 128, 0, stream>>>(rowp, colp, nrm, xs[i], agg);
        k_gemm_fused<<<ggrid, T, 0, stream>>>(agg, xs[i], W, Wr,
                                              out + (size_t)i * Nn * D, Nn);
    }
}